// CustomPenaltyLayer_6640019440151
// MI455X (gfx1250) — compile-verified
//
#include <hip/hip_runtime.h>
#include <hip/hip_bf16.h>

// ---------------------------------------------------------------------------
// Penalty reduction over x:(1024,4096,8) f32 using columns 2 and 3.
// Memory-bound: one streaming pass (~134 MB @ 23.3 TB/s ~ 5.8us floor).
// Wave owns 512 contiguous rows; lane-31 wrap transition comes from
// v_readfirstlane of the next iteration (no extra loads/divides).
// Wave-level reduction uses V_WMMA_F32_16X16X4_F32 (ones-matrix reduce).
// ---------------------------------------------------------------------------

typedef __attribute__((ext_vector_type(2))) float v2f;
typedef __attribute__((ext_vector_type(8))) float v8f;

#define NROWS          (1024 * 4096)       // 4,194,304
#define BLOCK          256                 // 8 waves of 32
#define RPT            16                  // iterations (rows per lane)
#define ROWS_PER_WAVE  (32 * RPT)          // 512 contiguous rows per wave
#define ROWS_PER_BLOCK (BLOCK * RPT)       // 4096
#define NBLOCKS        (NROWS / ROWS_PER_BLOCK)  // 1024

// 32-lane sum via WMMA: A VGPR0 = v, VGPR1 = 0; B = all-ones (layout
// invariant) -> D[m][n] = rowsum(m). Sum of a lane's 8 D-VGPRs covers 8 rows;
// lanes n and n^16 cover all 16 rows of one column; shfl_xor(16) finishes.
// All values are small non-negative integers -> every f32 add is exact.
__device__ __forceinline__ float wave_sum_wmma(float v) {
    v2f a; a.x = v;    a.y = 0.0f;
    v2f b; b.x = 1.0f; b.y = 1.0f;
    v8f c = {};
    v8f d = __builtin_amdgcn_wmma_f32_16x16x4_f32(
        false, a, false, b, (short)0, c, false, false);
    float s = d[0] + d[1] + d[2] + d[3] + d[4] + d[5] + d[6] + d[7];
    s += __shfl_xor(s, 16, 32);
    return s;
}

// invalid-transition contribution for pair (p = a[i], c = a[i+1]).
// fmodf(p,2)==0  <=>  p is an even integer  <=>  floorf(0.5f*p) == 0.5f*p
// (0.5f*p exact). p >= -FLT_MAX guard makes p=-inf give false, matching
// jnp.mod(-inf,2)=NaN != 0. NaN p: floor(NaN)!=NaN -> false, matches.
__device__ __forceinline__ float trans_pen(float p, float c) {
    const float h = 0.5f * p;
    const bool cond    = (floorf(h) == h) && (p < 20.0f)
                         && (p >= -3.402823466e38f);
    const bool invalid = (c != p + 1.0f) && (c != 22.0f);
    return (cond && invalid) ? 1.0f : 0.0f;
}

__global__ __launch_bounds__(BLOCK)
void penalty_main(const float* __restrict__ x,
                  const float* __restrict__ mn,
                  const float* __restrict__ sc,
                  float2* __restrict__ partials) {
    const float m2 = mn[2], m3 = mn[3];
    const float s2 = sc[2], s3 = sc[3];

    const unsigned lane = threadIdx.x & 31u;
    const unsigned wave = threadIdx.x >> 5;
    const bool is31 = (lane == 31u);

    const size_t waveBase =
        ((size_t)blockIdx.x * (BLOCK / 32) + wave) * ROWS_PER_WAVE;
    // cols 2..3 of row (waveBase + lane): byte offset row*32 + 8 (8B aligned)
    const float2* __restrict__ pf =
        reinterpret_cast<const float2*>(x + (waveBase + lane) * 8 + 2);

    float pen = 0.0f;   // device + activity + transition penalties
    float cnt = 0.0f;   // num_activities (a != 22)
    float prev = 0.0f;  // lane 31's a from previous iteration

    #pragma unroll
    for (int k = 0; k < RPT; ++k) {
        // row = waveBase + k*32 + lane ; offset = k*32 rows = k*128 float2
        const float2 v = pf[(size_t)k * 128];
        const float d = (v.x - m2) / s2;
        const float a = (v.y - m3) / s3;

        // device penalty: where NOT (0 <= d <= 252)   (NaN -> penalty)
        pen += (d >= 0.0f && d <= 252.0f) ? 0.0f : 1.0f;
        // activity range penalty: (a < 0) | (a > 22)
        pen += (a < 0.0f || a > 22.0f) ? 1.0f : 0.0f;
        // activity count: a != 22
        cnt += (a != 22.0f) ? 1.0f : 0.0f;

        // successor values: lanes 0..30 -> neighbor lane; lane 31 pairs its
        // PREVIOUS iteration's a with this iteration's lane-0 a (row wrap).
        const float an = __shfl_down(a, 1, 32);
        const float a0 = __uint_as_float(
            __builtin_amdgcn_readfirstlane(__float_as_uint(a)));

        const float P = is31 ? prev : a;
        const float C = is31 ? a0   : an;
        float tp = trans_pen(P, C);
        if (k == 0) tp = is31 ? 0.0f : tp;   // no wrap pair on first iter
        pen += tp;

        prev = a;
    }

    // tail pair (waveBase+511, waveBase+512): one load+divide per wave
    const size_t succ = waveBase + ROWS_PER_WAVE;
    if (is31 && succ < (size_t)NROWS) {
        const float asucc = (x[succ * 8 + 3] - m3) / s3;
        pen += trans_pen(prev, asucc);
    }

    // wave reduce via WMMA (EXEC all ones: full block, converged here)
    const float wpen = wave_sum_wmma(pen);
    const float wcnt = wave_sum_wmma(cnt);

    __shared__ float2 sdata[BLOCK / 32];
    if (lane == 0) {
        sdata[wave] = make_float2(wpen, wcnt);
    }
    __syncthreads();
    if (threadIdx.x == 0) {
        float bp = 0.0f, bc = 0.0f;
        #pragma unroll
        for (int w = 0; w < BLOCK / 32; ++w) { bp += sdata[w].x; bc += sdata[w].y; }
        partials[blockIdx.x] = make_float2(bp, bc);
    }
}

__global__ __launch_bounds__(256)
void penalty_final(const float2* __restrict__ partials, float* __restrict__ out) {
    __shared__ float2 sh[256];
    float p = 0.0f, c = 0.0f;
    for (int i = threadIdx.x; i < NBLOCKS; i += 256) {
        const float2 v = partials[i];
        p += v.x;
        c += v.y;
    }
    sh[threadIdx.x] = make_float2(p, c);
    __syncthreads();
    #pragma unroll
    for (int s = 128; s > 0; s >>= 1) {
        if ((int)threadIdx.x < s) {
            sh[threadIdx.x].x += sh[threadIdx.x + s].x;
            sh[threadIdx.x].y += sh[threadIdx.x + s].y;
        }
        __syncthreads();
    }
    if (threadIdx.x == 0) {
        out[0] = sh[0].x + fabsf(sh[0].y - 58.0f);
    }
}

extern "C" void kernel_launch(void* const* d_in, const int* in_sizes, int n_in,
                              void* d_out, int out_size, void* d_ws, size_t ws_size,
                              hipStream_t stream) {
    (void)in_sizes; (void)n_in; (void)out_size; (void)ws_size;
    const float* x  = (const float*)d_in[0];
    const float* mn = (const float*)d_in[1];
    const float* sc = (const float*)d_in[2];
    float2* partials = (float2*)d_ws;            // NBLOCKS * 8 bytes = 8 KB
    float*  out      = (float*)d_out;

    penalty_main <<<NBLOCKS, BLOCK, 0, stream>>>(x, mn, sc, partials);
    penalty_final<<<1, 256, 0, stream>>>(partials, out);
}